// Qwen3NextGatedDeltaNetDecode_62337155334415
// MI455X (gfx1250) — compile-verified
//
#include <hip/hip_runtime.h>

// ---------------------------------------------------------------------------
// Qwen3-Next GatedDeltaNet single-token decode, fused, MI455X (gfx1250).
//
// o = eg*(S@qn) + beta*(v - eg*(S@kn)) * (kn.qn)      (S not written back)
//
// Bandwidth bound: 512 MiB of ssm_state streamed exactly once (~22us @23.3TB/s).
// Dual matvec expressed with V_WMMA_F32_16X16X4_F32:
//   A (16x4)  : row0 = kn chunk, row1 = qn chunk, rows 2..15 = 0 (LDS)
//   B (4x16)  : S tile, N = v-row; lane l holds S[vb+(l&15), k+2*(l>>4)+j]
//   C (16x16) : M=0 row -> S@kn, M=1 row -> S@qn  (acc[0], acc[1], lanes 0-15)
// One wave32 per (batch, v-head) pair: 8192 waves.
// ---------------------------------------------------------------------------

namespace {

constexpr int HK    = 128;
constexpr int HV    = 128;
constexpr int NK    = 16;
constexpr int NV    = 32;
constexpr int QKV   = 2 * NK * HK + NV * HV;   // 8192
constexpr int BATCH = 256;
constexpr int WAVES = 8;                        // waves per block
constexpr int PAIRS = BATCH * NV;               // 8192

typedef float v2f __attribute__((ext_vector_type(2)));
typedef float v8f __attribute__((ext_vector_type(8)));

__device__ __forceinline__ void conv_silu4(
    const float* __restrict__ x, const float* __restrict__ cs,
    const float* __restrict__ cw, int bb, int cbase, float out[4])
{
  const float* xr = x  + (size_t)bb * QKV + cbase;
  const float* s0 = cs + ((size_t)bb * 3 + 0) * QKV + cbase;
  const float* s1 = cs + ((size_t)bb * 3 + 1) * QKV + cbase;
  const float* s2 = cs + ((size_t)bb * 3 + 2) * QKV + cbase;
  float xx[4], a0[4], a1[4], a2[4];
  *(float4*)xx = *(const float4*)xr;
  *(float4*)a0 = *(const float4*)s0;
  *(float4*)a1 = *(const float4*)s1;
  *(float4*)a2 = *(const float4*)s2;
#pragma unroll
  for (int j = 0; j < 4; ++j) {
    const float4 w = *(const float4*)(cw + (size_t)(cbase + j) * 4);
    const float co = a0[j] * w.x + a1[j] * w.y + a2[j] * w.z + xx[j] * w.w;
    out[j] = co / (1.0f + __expf(-co));   // SiLU
  }
}

__device__ __forceinline__ float wave_sum(float v) {
#pragma unroll
  for (int off = 16; off > 0; off >>= 1)
    v += __shfl_xor(v, off, 32);
  return v;
}

} // namespace

__global__ __launch_bounds__(WAVES * 32) void gdn_decode(
    const float* __restrict__ mixed_qkv,
    const float* __restrict__ b_in,
    const float* __restrict__ a_in,
    const float* __restrict__ conv_state,
    const float* __restrict__ conv_w,
    const float* __restrict__ ssm_state,
    const float* __restrict__ alog,
    const float* __restrict__ dt_bias,
    float* __restrict__ out)
{
  __shared__ float lds[WAVES * 640];
  const int tid  = threadIdx.x;
  const int wv   = tid >> 5;
  const int lane = tid & 31;
  const int pair = blockIdx.x * WAVES + wv;   // (b, h) flat, 0..8191
  const int bb   = pair >> 5;                 // batch
  const int h    = pair & 31;                 // v-head
  const int kh   = h >> 1;                    // shared q/k head (GROUP=2)

  float* wl     = lds + wv * 640;
  float* lds_kn = wl;            // [128]
  float* lds_qn = wl + 128;      // [128]
  float* lds_z  = wl + 256;      // [128] zeros (A rows 2..15)
  float* lds_u  = wl + 384;      // [128] S @ kn
  float* lds_w  = wl + 512;      // [128] S @ qn

  // ---- causal conv update + SiLU for this wave's q/k/v channels ----
  float qr[4], kr[4], vr[4];
  conv_silu4(mixed_qkv, conv_state, conv_w, bb, kh * HK + 4 * lane, qr);
  conv_silu4(mixed_qkv, conv_state, conv_w, bb, NK * HK + kh * HK + 4 * lane, kr);
  conv_silu4(mixed_qkv, conv_state, conv_w, bb, 2 * NK * HK + h * HV + 4 * lane, vr);

  // ---- L2 norms (eps inside sqrt), q scaled by HK^-0.5, and kn.qn ----
  float sq = qr[0]*qr[0] + qr[1]*qr[1] + qr[2]*qr[2] + qr[3]*qr[3];
  float sk = kr[0]*kr[0] + kr[1]*kr[1] + kr[2]*kr[2] + kr[3]*kr[3];
  sq = wave_sum(sq);
  sk = wave_sum(sk);
  const float qs = rsqrtf(sq + 1e-6f) * 0.08838834764831845f;  // * 128^-0.5
  const float ks = rsqrtf(sk + 1e-6f);
  float qn[4], kn[4], pdot = 0.f;
#pragma unroll
  for (int j = 0; j < 4; ++j) {
    qn[j] = qr[j] * qs;
    kn[j] = kr[j] * ks;
    pdot += qn[j] * kn[j];
  }
  const float kq = wave_sum(pdot);

#pragma unroll
  for (int j = 0; j < 4; ++j) {
    lds_kn[4 * lane + j] = kn[j];
    lds_qn[4 * lane + j] = qn[j];
    lds_z [4 * lane + j] = 0.f;
  }
  __syncthreads();   // uniform across all waves in the block

  // ---- gating: g = -exp(alog)*softplus(a+dt_bias); eg = exp(g); beta = sigm(b)
  const float av = a_in[bb * NV + h] + dt_bias[h];
  const float sp = fmaxf(av, 0.f) + log1pf(__expf(-fabsf(av)));  // stable softplus
  const float eg = __expf(-__expf(alog[h]) * sp);
  const float beta = 1.f / (1.f + __expf(-b_in[bb * NV + h]));

  // ---- WMMA dual matvec: stream the 64 KiB S slice once ----
  const int half = lane >> 4;
  const int m    = lane & 15;
  // A operand (16x4 f32): lane l holds A[l&15, 2*(l>>4)+{0,1}]
  const float* Ap = ((m == 0) ? lds_kn : (m == 1) ? lds_qn : lds_z) + 2 * half;
  // B operand (4x16 f32): lane l holds B[2*(l>>4)+{0,1}, l&15] = S[vb+m, k+...]
  const float* Bp = ssm_state + ((size_t)pair * HV + m) * (size_t)HK + 2 * half;

  for (int mt = 0; mt < 8; ++mt) {      // 8 tiles of 16 v-rows
    v8f acc = {0.f, 0.f, 0.f, 0.f, 0.f, 0.f, 0.f, 0.f};
    const float* Bt = Bp + mt * 16 * HK;
#pragma unroll
    for (int kc = 0; kc < 32; ++kc) {   // K=128 in chunks of 4
      const v2f a = *(const v2f*)(Ap + kc * 4);
      const v2f s = *(const v2f*)(Bt + kc * 4);
      acc = __builtin_amdgcn_wmma_f32_16x16x4_f32(
          false, a, false, s, (short)0, acc, false, false);
    }
    // C: VGPR r, lanes 0-15 -> M=r, N=lane. M=0 row = S@kn, M=1 row = S@qn.
    if (half == 0) {
      lds_u[mt * 16 + m] = acc[0];
      lds_w[mt * 16 + m] = acc[1];
    }
  }
  __syncthreads();

  // ---- epilogue: o = eg*w + beta*(v - eg*u)*kq ; coalesced float4 store ----
  float o[4];
#pragma unroll
  for (int j = 0; j < 4; ++j) {
    const int vi = 4 * lane + j;
    o[j] = eg * lds_w[vi] + beta * (vr[j] - eg * lds_u[vi]) * kq;
  }
  *(float4*)(out + (size_t)pair * HV + 4 * lane) = make_float4(o[0], o[1], o[2], o[3]);
}

extern "C" void kernel_launch(void* const* d_in, const int* in_sizes, int n_in,
                              void* d_out, int out_size, void* d_ws, size_t ws_size,
                              hipStream_t stream) {
  (void)in_sizes; (void)n_in; (void)out_size; (void)d_ws; (void)ws_size;
  const float* mixed_qkv  = (const float*)d_in[0];
  const float* b_in       = (const float*)d_in[1];
  const float* a_in       = (const float*)d_in[2];
  const float* conv_state = (const float*)d_in[3];
  const float* conv_w     = (const float*)d_in[4];
  const float* ssm_state  = (const float*)d_in[5];
  const float* alog       = (const float*)d_in[6];
  const float* dt_bias    = (const float*)d_in[7];
  float* out = (float*)d_out;

  dim3 grid(PAIRS / WAVES);     // 1024 blocks
  dim3 block(WAVES * 32);       // 256 threads = 8 wave32
  hipLaunchKernelGGL(gdn_decode, grid, block, 0, stream,
      mixed_qkv, b_in, a_in, conv_state, conv_w, ssm_state, alog, dt_bias, out);
}